// TypeSpecificNet_22393959481888
// MI455X (gfx1250) — compile-verified
//
#include <hip/hip_runtime.h>
#include <hip/hip_bf16.h>
#include <math.h>

// Problem sizes (fixed by the reference)
#define BB 4096
#define DD 256
#define CC 64
#define NORM_EPS 1e-10f

typedef __attribute__((ext_vector_type(2))) float v2f;
typedef __attribute__((ext_vector_type(8))) float v8f;

__device__ __forceinline__ float waveReduceSum(float v) {
    v += __shfl_xor(v, 1, 32);
    v += __shfl_xor(v, 2, 32);
    v += __shfl_xor(v, 4, 32);
    v += __shfl_xor(v, 8, 32);
    v += __shfl_xor(v, 16, 32);
    return v;
}

// ---------------- K0: zero the small ws accumulators ----------------
__global__ void tsn_init_kernel(float* l1, float* sumsq, int* counts, int* cursor) {
    int t = threadIdx.x;
    if (t < CC) { l1[t] = 0.0f; counts[t] = 0; cursor[t] = 0; }
    if (t == 0) { *sumsq = 0.0f; }
}

// ---------------- K1: per-condition L1 norm of W ----------------
// grid (C, 16), block 256; each block reduces 4096 elements of one condition.
__global__ __launch_bounds__(256) void tsn_condl1_kernel(const float* __restrict__ W,
                                                         float* __restrict__ l1) {
    const int cond = blockIdx.x;
    const float* Wc = W + (size_t)cond * DD * DD;
    const int base = blockIdx.y * 4096 + threadIdx.x;
    float s = 0.0f;
#pragma unroll
    for (int i = 0; i < 16; ++i) s += fabsf(Wc[base + i * 256]);
    s = waveReduceSum(s);
    if ((threadIdx.x & 31) == 0) atomicAdd(&l1[cond], s);
}

// ---------------- K2: histogram of condition indices ----------------
__global__ void tsn_hist_kernel(const int* __restrict__ c, int* __restrict__ counts) {
    int i = blockIdx.x * blockDim.x + threadIdx.x;
    if (i < BB) atomicAdd(&counts[c[i]], 1);
}

// ---------------- K3: copy x to output tail + sum of squares ----------------
// grid 1024, block 256, 4 elements/thread covers B*D = 1,048,576 exactly.
__global__ __launch_bounds__(256) void tsn_xpass_kernel(const float* __restrict__ x,
                                                        float* __restrict__ out_x,
                                                        float* __restrict__ sumsq) {
    const int base = (blockIdx.x * blockDim.x + threadIdx.x) * 4;
    float s = 0.0f;
#pragma unroll
    for (int i = 0; i < 4; ++i) {
        float v = x[base + i];
        out_x[base + i] = v;
        s += v * v;
    }
    s = waveReduceSum(s);
    if ((threadIdx.x & 31) == 0) atomicAdd(sumsq, s);
}

// ---------------- K4: offsets prefix-sum, mask_norm, embed_norm ----------------
__global__ void tsn_finalize_kernel(const int* __restrict__ counts, const float* __restrict__ l1,
                                    const float* __restrict__ sumsq, int* __restrict__ offsets,
                                    float* __restrict__ out_scalars) {
    if (threadIdx.x == 0) {
        int acc = 0;
        float mn = 0.0f;
        for (int i = 0; i < CC; ++i) {
            offsets[i] = acc;
            acc += counts[i];
            mn += (float)counts[i] * l1[i];
        }
        out_scalars[0] = mn;              // mask_norm
        out_scalars[1] = sqrtf(*sumsq);   // embed_norm
    }
}

// ---------------- K5: counting-sort scatter of sample ids by condition ----------------
__global__ void tsn_scatter_kernel(const int* __restrict__ c, const int* __restrict__ offsets,
                                   int* __restrict__ cursor, int* __restrict__ idx_sorted) {
    int i = blockIdx.x * blockDim.x + threadIdx.x;
    if (i < BB) {
        int cond = c[i];
        int pos = offsets[cond] + atomicAdd(&cursor[cond], 1);
        idx_sorted[pos] = i;
    }
}

// ---------------- K6: grouped GEMM + bias + row-normalize via WMMA f32 16x16x4 ----------------
// grid (C, TILE_SLOTS); block = 256 threads (8 waves). One block iteration handles a
// 16-sample tile of one condition; 8 waves x 2 N-tiles cover all 256 output columns.
// Epilogue goes through an LDS y-tile: conflict-free stores, ds_load_b128 row-sumsq with
// fire-and-forget ds_add_f32 atomics, then fully coalesced normalized global stores.
__global__ __launch_bounds__(256) void tsn_gemm_kernel(const float* __restrict__ x,
                                                       const float* __restrict__ W,
                                                       const float* __restrict__ bias,
                                                       const int* __restrict__ idx_sorted,
                                                       const int* __restrict__ counts,
                                                       const int* __restrict__ offsets,
                                                       float* __restrict__ out_me) {
    __shared__ __align__(16) float xs[16][DD];   // 16 KB staged X tile
    __shared__ __align__(16) float ys[16][DD];   // 16 KB y tile (pre-normalization)
    __shared__ float rowsq[16];
    __shared__ int   srows[16];

    const int cond = blockIdx.x;
    const int cnt  = counts[cond];
    const int base = offsets[cond];
    const int tid  = threadIdx.x;
    const int wave = tid >> 5;           // 0..7
    const int lane = tid & 31;
    const int half = lane >> 4;          // 0: lanes 0-15, 1: lanes 16-31
    const int l16  = lane & 15;
    const float* Wc = W + (size_t)cond * DD * DD;

    for (int tile = blockIdx.y; tile * 16 < cnt; tile += gridDim.y) {
        if (tid < 16) {
            int r = tile * 16 + tid;
            srows[tid] = (r < cnt) ? idx_sorted[base + r] : -1;
            rowsq[tid] = 0.0f;
        }
        __syncthreads();

        // Stage X rows (coalesced); padded rows -> 0
        for (int e = tid; e < 16 * DD; e += 256) {
            int r = e >> 8, col = e & 255;
            int s = srows[r];
            xs[r][col] = (s >= 0) ? x[(size_t)s * DD + col] : 0.0f;
        }
        __syncthreads();

#pragma unroll
        for (int nt = 0; nt < 2; ++nt) {
            const int n0 = (wave * 2 + nt) * 16;
            v8f acc = {0.f, 0.f, 0.f, 0.f, 0.f, 0.f, 0.f, 0.f};
            // D = A(16x4) * B(4x16) + C, K stepped by 4 over 256.
            // A frag (ISA layout): lane l16 = M row; VGPR j holds K = k0 + 2*half + j
            // B frag (mirrored):   lane l16 = N col; VGPR j holds K = k0 + 2*half + j
            // B[k][n] = W[n][k]  (y[m,n] = sum_k X[m,k]*W[n,k])
            for (int k0 = 0; k0 < DD; k0 += 4) {
                const int ka = k0 + 2 * half;
                v2f a = *(const v2f*)(&xs[l16][ka]);
                v2f b = *(const v2f*)(Wc + (size_t)(n0 + l16) * DD + ka);
                acc = __builtin_amdgcn_wmma_f32_16x16x4_f32(
                    /*neg_a=*/false, a, /*neg_b=*/false, b,
                    /*c_mod=*/(short)0, acc, /*reuse_a=*/false, /*reuse_b=*/false);
            }
            // C/D layout: VGPR v, lane<16 -> (M=v, N=n0+l16); lane>=16 -> (M=v+8, N=n0+l16)
            float bv = bias[cond * DD + n0 + l16];
#pragma unroll
            for (int v = 0; v < 8; ++v) ys[v + 8 * half][n0 + l16] = acc[v] + bv;
        }
        __syncthreads();  // y tile complete in LDS

        // Row sum of squares: 16 threads per row, b128 reads, one LDS float atomic each
        {
            const int r = tid >> 4, seg = tid & 15;
            const float4* p = (const float4*)&ys[r][seg * 16];
            float s = 0.0f;
#pragma unroll
            for (int q = 0; q < 4; ++q) {
                float4 v4 = p[q];
                s += v4.x * v4.x + v4.y * v4.y + v4.z * v4.z + v4.w * v4.w;
            }
            atomicAdd(&rowsq[r], s);
        }
        __syncthreads();  // rowsq complete

        // Normalize + coalesced scatter: per row, 256 lanes store 1 KB contiguous
        for (int rr = 0; rr < 16; ++rr) {
            int s = srows[rr];
            if (s >= 0) {
                float inv = 1.0f / fmaxf(sqrtf(rowsq[rr]), NORM_EPS);
                out_me[(size_t)s * DD + tid] = ys[rr][tid] * inv;
            }
        }
        __syncthreads();  // before LDS reuse in next persistent tile
    }
}

extern "C" void kernel_launch(void* const* d_in, const int* in_sizes, int n_in,
                              void* d_out, int out_size, void* d_ws, size_t ws_size,
                              hipStream_t stream) {
    (void)in_sizes; (void)n_in; (void)out_size; (void)ws_size;
    const float* x    = (const float*)d_in[0];   // [B, D]
    const float* W    = (const float*)d_in[1];   // [C, D, D]
    const float* bvec = (const float*)d_in[2];   // [C, D]
    const int*   c    = (const int*)d_in[3];     // [B]
    float* out = (float*)d_out;

    // ws layout (byte offsets): l1[64]f @0, sumsq f @256, counts[64]i @272,
    // offsets[64]i @528, cursor[64]i @784, idx_sorted[4096]i @1040  (~17.4 KB)
    char* ws = (char*)d_ws;
    float* l1         = (float*)(ws + 0);
    float* sumsq      = (float*)(ws + 256);
    int*   counts     = (int*)(ws + 272);
    int*   offsets    = (int*)(ws + 528);
    int*   cursor     = (int*)(ws + 784);
    int*   idx_sorted = (int*)(ws + 1040);

    float* out_me      = out;                          // [B*D] masked_embedding
    float* out_scalars = out + (size_t)BB * DD;        // mask_norm, embed_norm
    float* out_x       = out + (size_t)BB * DD + 2;    // [B*D] x passthrough

    tsn_init_kernel<<<1, 256, 0, stream>>>(l1, sumsq, counts, cursor);
    tsn_condl1_kernel<<<dim3(CC, 16), 256, 0, stream>>>(W, l1);
    tsn_hist_kernel<<<BB / 256, 256, 0, stream>>>(c, counts);
    tsn_xpass_kernel<<<(BB * DD) / (256 * 4), 256, 0, stream>>>(x, out_x, sumsq);
    tsn_finalize_kernel<<<1, 1, 0, stream>>>(counts, l1, sumsq, offsets, out_scalars);
    tsn_scatter_kernel<<<BB / 256, 256, 0, stream>>>(c, offsets, cursor, idx_sorted);
    tsn_gemm_kernel<<<dim3(CC, 16), 256, 0, stream>>>(x, W, bvec, idx_sorted, counts,
                                                      offsets, out_me);
}